// QuantileLoss_14834817040526
// MI455X (gfx1250) — compile-verified
//
#include <hip/hip_runtime.h>

// CDNA5 / gfx1250, wave32. Bandwidth-bound streaming reduction (~335 MB read,
// roofline ~14.4 us @ 23.3 TB/s). Epilogue wave reduction done with
// V_WMMA_F32_16X16X4_F32 (ones-matrix trick): 64 lane partials -> 16 row sums
// in one matrix op.

typedef __attribute__((ext_vector_type(2))) float v2f;
typedef __attribute__((ext_vector_type(8))) float v8f;

__device__ __forceinline__ float quantile_terms(float pmid, float plo, float phi, float tt) {
    const float PEN = 1000.0f;
    float dlo = plo - tt;
    float dhi = phi - tt;
    float lower = (plo > pmid) ? PEN : ((plo > tt * 0.95f) ? 0.0f : dlo * dlo);
    float upper = (phi < pmid) ? PEN : ((phi < tt * 1.05f) ? 0.0f : dhi * dhi);
    return lower + upper;
}

__global__ void __launch_bounds__(256)
quantile_loss_kernel(const float* __restrict__ preds,
                     const float* __restrict__ target,
                     float* __restrict__ out,
                     long long pairs,      // N/2
                     long long n_rows,     // N
                     float scale)          // 1 / (5*N)
{
    const float4* __restrict__ P = reinterpret_cast<const float4*>(preds);
    const float4* __restrict__ T = reinterpret_cast<const float4*>(target);

    const long long tid    = (long long)blockIdx.x * blockDim.x + threadIdx.x;
    const long long stride = (long long)gridDim.x * blockDim.x;

    // Two accumulators per lane: together a wave's 64 partials fill the
    // 16x4 f32 WMMA A-matrix exactly.
    float acc0 = 0.0f;
    float acc1 = 0.0f;

    for (long long i = tid; i < pairs; i += stride) {
        // Prefetch next grid-stride chunk (speculative; OOB silently dropped).
        __builtin_prefetch(&P[3 * (i + stride)], 0, 3);
        __builtin_prefetch(&T[2 * (i + stride)], 0, 3);

        // 2 rows of preds = 3 x b128, 2 rows of target = 2 x b128 (16B aligned).
        float4 q0 = P[3 * i + 0];   // row0: p0 p1 p2 p3
        float4 q1 = P[3 * i + 1];   // row0: p4 p5 | row1: p0 p1
        float4 q2 = P[3 * i + 2];   // row1: p2 p3 p4 p5
        float4 u0 = T[2 * i + 0];   // row0: t0 t1 t2 t3
        float4 u1 = T[2 * i + 1];   // row1: t0 t1 t2 t3

        // Row 0
        float d0 = q0.x - u0.x, d1 = q0.y - u0.y, d2 = q0.z - u0.z;
        acc0 += d0 * d0 + d1 * d1 + d2 * d2
              + quantile_terms(q0.w, q1.x, q1.y, u0.w);

        // Row 1
        float e0 = q1.z - u1.x, e1 = q1.w - u1.y, e2 = q2.x - u1.z;
        acc1 += e0 * e0 + e1 * e1 + e2 * e2
              + quantile_terms(q2.y, q2.z, q2.w, u1.w);
    }

    // Odd-row tail (N fixed even in reference; kept for generality).
    if (tid == 0 && (n_rows & 1LL)) {
        long long r = n_rows - 1;
        const float* p = preds + 6 * r;
        const float* t = target + 4 * r;
        float d0 = p[0] - t[0], d1 = p[1] - t[1], d2 = p[2] - t[2];
        acc0 += d0 * d0 + d1 * d1 + d2 * d2
              + quantile_terms(p[3], p[4], p[5], t[3]);
    }

    // ---- Wave32 reduction via V_WMMA_F32_16X16X4_F32 (ones-matrix trick) ----
    // A (16x4): lane L<16 holds A[L][0],A[L][1]; lane L+16 holds A[L][2],A[L][3].
    // B (4x16) = all ones  =>  D[m][n] = sum_k A[m][k]  (16 row-sums, replicated
    // across columns). EXEC is all 1s here: full 256-thread blocks, reconverged
    // after the loop.
    v2f a; a.x = acc0; a.y = acc1;
    v2f b; b.x = 1.0f; b.y = 1.0f;
    v8f c = {};
    c = __builtin_amdgcn_wmma_f32_16x16x4_f32(
            /*neg_a=*/false, a, /*neg_b=*/false, b,
            /*c_mod=*/(short)0, c, /*reuse_a=*/false, /*reuse_b=*/false);

    // D layout: VGPR v, lanes 0-15 = row v, lanes 16-31 = row v+8.
    // Lane-local sum over 8 VGPRs -> half-wave sum; xor-16 swap -> wave sum.
    float t = c[0] + c[1] + c[2] + c[3] + c[4] + c[5] + c[6] + c[7];
    t += __shfl_xor(t, 16, 32);

    // ---- Cross-wave (8 waves / 256-thread block) + cross-block reduction ----
    __shared__ float wsum[8];
    const int lane = threadIdx.x & 31;
    const int wave = threadIdx.x >> 5;
    if (lane == 0) wsum[wave] = t;
    __syncthreads();
    if (threadIdx.x == 0) {
        float s = 0.0f;
        const int nw = (blockDim.x + 31) >> 5;
        for (int w = 0; w < nw; ++w) s += wsum[w];
        atomicAdd(out, s * scale);   // global_atomic_add_f32, pre-scaled by 1/(5N)
    }
}

extern "C" void kernel_launch(void* const* d_in, const int* in_sizes, int n_in,
                              void* d_out, int out_size, void* d_ws, size_t ws_size,
                              hipStream_t stream) {
    const float* preds  = (const float*)d_in[0];   // (N, 6) f32
    const float* target = (const float*)d_in[1];   // (N, 4) f32
    float* out = (float*)d_out;                    // scalar f32

    const long long n_rows = (long long)in_sizes[0] / 6;   // N
    const long long pairs  = n_rows / 2;
    const float scale = (float)(1.0 / (5.0 * (double)n_rows));

    // d_out is poisoned by the harness; zero it (graph-capturable memset node).
    hipMemsetAsync(out, 0, sizeof(float), stream);

    const int threads = 256;           // 8 waves per WGP-resident block
    const int blocks  = 2048;          // 524,288 threads -> exactly 8 pairs/thread
    quantile_loss_kernel<<<blocks, threads, 0, stream>>>(
        preds, target, out, pairs, n_rows, scale);
}